// FairnessAwareContrastiveLoss_3891240370210
// MI455X (gfx1250) — compile-verified
//
#include <hip/hip_runtime.h>
#include <hip/hip_bf16.h>
#include <math.h>

#define ALPHA_C 0.65f
#define BETA_C  0.35f
#define MARGIN_C 1.0f
#define EPS_C   1e-8f
#define DIM 512

typedef float v2f __attribute__((ext_vector_type(2)));
typedef float v8f __attribute__((ext_vector_type(8)));

// ---------------- zero the accumulator region of workspace ----------------
__global__ void facl_zero_ws(float* __restrict__ ws, int n) {
    int i = blockIdx.x * blockDim.x + threadIdx.x;
    if (i < n) ws[i] = 0.0f;
}

// ---------------- phase 1: per-16-row block Gram diagonals via WMMA -------
// One wave (32 lanes) per 16-row block. EXEC is all-ones for every WMMA.
__global__ __launch_bounds__(32)
void facl_phase1(const float* __restrict__ R, const float* __restrict__ S,
                 float* __restrict__ u, float* __restrict__ v,
                 float* __restrict__ pos,
                 float* __restrict__ sum_pos, float* __restrict__ sum_fair) {
    const int lane = threadIdx.x;
    const int row0 = blockIdx.x * 16;
    __shared__ float inr[16], ins[16];

    float drr = 0.f, dss = 0.f, drs = 0.f;
    int md = -1;

#if __has_builtin(__builtin_amdgcn_wmma_f32_16x16x4_f32)
    // A-matrix 16x4 f32 layout: lanes 0-15 -> M=0..15 holding {K=0,K=1};
    // lanes 16-31 -> same M holding {K=2,K=3}. B loaded identically from
    // row-major S gives B[k][n] = S[n][k], i.e. C = R * S^T.
    const int m    = lane & 15;
    const int koff = (lane >> 4) * 2;
    const float* rrow = R + (size_t)(row0 + m) * DIM + koff;
    const float* srow = S + (size_t)(row0 + m) * DIM + koff;

    v8f crr = {}, css = {}, crs = {};
    #pragma unroll 4
    for (int k = 0; k < DIM; k += 4) {
        v2f a = *(const v2f*)(rrow + k);
        v2f b = *(const v2f*)(srow + k);
        crr = __builtin_amdgcn_wmma_f32_16x16x4_f32(false, a, false, a,
                                                    (short)0, crr, false, false);
        css = __builtin_amdgcn_wmma_f32_16x16x4_f32(false, b, false, b,
                                                    (short)0, css, false, false);
        crs = __builtin_amdgcn_wmma_f32_16x16x4_f32(false, a, false, b,
                                                    (short)0, crs, false, false);
    }
    // C/D layout: VGPR g, lanes 0-15 -> (M=g, N=lane); lanes 16-31 -> (M=g+8, N=lane-16).
    // Diagonal (m,m): m<8 at VGPR m / lane m ; m>=8 at VGPR m-8 / lane m+16.
    #pragma unroll
    for (int g = 0; g < 8; ++g) {
        if (lane == g)      { drr = crr[g]; dss = css[g]; drs = crs[g]; md = g;     }
        if (lane == g + 24) { drr = crr[g]; dss = css[g]; drs = crs[g]; md = g + 8; }
    }
#else
    if (lane < 16) {
        md = lane;
        const float* rrow = R + (size_t)(row0 + lane) * DIM;
        const float* srow = S + (size_t)(row0 + lane) * DIM;
        for (int d = 0; d < DIM; ++d) {
            float a = rrow[d], b = srow[d];
            drr = fmaf(a, a, drr); dss = fmaf(b, b, dss); drs = fmaf(a, b, drs);
        }
    }
#endif

    if (md >= 0) {
        const int row = row0 + md;
        float nr = sqrtf(drr), ns = sqrtf(dss);
        float p  = drs / fmaxf(nr * ns, EPS_C);
        pos[row] = p;
        float pc   = fminf(fmaxf(p, -1.0f), 1.0f);
        float ang  = acosf(pc);
        float eu   = sqrtf(fmaxf(drr + dss - 2.0f * drs, 0.0f));
        float comb = 2.0f * eu * ang / (eu + ang + EPS_C);
        atomicAdd(sum_fair, comb);
        atomicAdd(sum_pos, p);
        inr[md] = 1.0f / nr;
        ins[md] = 1.0f / ns;
    }
    __syncthreads();

    // Coalesced second pass: block-partial column sums of r_i/||r_i||, s_j/||s_j||.
    for (int j = 0; j < DIM / 32; ++j) {
        const int d = lane + 32 * j;
        float au = 0.f, av = 0.f;
        #pragma unroll
        for (int m2 = 0; m2 < 16; ++m2) {
            au = fmaf(R[(size_t)(row0 + m2) * DIM + d], inr[m2], au);
            av = fmaf(S[(size_t)(row0 + m2) * DIM + d], ins[m2], av);
        }
        atomicAdd(&u[d], au);
        atomicAdd(&v[d], av);
    }
}

// ---------------- phase 2: u.v, hinge fold, final scalar ------------------
__global__ __launch_bounds__(256)
void facl_phase2(const float* __restrict__ u, const float* __restrict__ v,
                 const float* __restrict__ pos,
                 const float* __restrict__ sum_pos, const float* __restrict__ sum_fair,
                 float* __restrict__ out, int N) {
    __shared__ float red[256];
    __shared__ float total_sh;
    const int t = threadIdx.x;

    float part = 0.f;
    for (int d = t; d < DIM; d += 256) part = fmaf(u[d], v[d], part);
    red[t] = part;
    __syncthreads();
    for (int s2 = 128; s2 > 0; s2 >>= 1) {
        if (t < s2) red[t] += red[t + s2];
        __syncthreads();
    }
    if (t == 0) total_sh = red[0];
    __syncthreads();

    const float neg = (total_sh - *sum_pos) / ((float)N * (float)(N - 1));

    float h = 0.f;
    for (int i = t; i < N; i += 256)
        h += fmaxf(MARGIN_C - pos[i] + neg, 0.0f);
    red[t] = h;
    __syncthreads();
    for (int s2 = 128; s2 > 0; s2 >>= 1) {
        if (t < s2) red[t] += red[t + s2];
        __syncthreads();
    }
    if (t == 0) {
        float contrastive = red[0] / (float)N;
        float fair        = (*sum_fair) / (float)N;
        *out = BETA_C * contrastive + ALPHA_C * fair;
    }
}

// --------------------------------------------------------------------------
extern "C" void kernel_launch(void* const* d_in, const int* in_sizes, int n_in,
                              void* d_out, int out_size, void* d_ws, size_t ws_size,
                              hipStream_t stream) {
    const float* R = (const float*)d_in[0];
    const float* S = (const float*)d_in[1];
    const int N = in_sizes[0] / DIM;   // 4096

    // ws layout (floats): [0]=sum_pos, [1]=sum_fair, [2..2+D)=u, [2+D..2+2D)=v,
    //                     [2+2D .. 2+2D+N) = pos
    float* ws       = (float*)d_ws;
    float* sum_pos  = ws;
    float* sum_fair = ws + 1;
    float* u        = ws + 2;
    float* v        = ws + 2 + DIM;
    float* pos      = ws + 2 + 2 * DIM;

    const int nzero = 2 + 2 * DIM;
    facl_zero_ws<<<(nzero + 255) / 256, 256, 0, stream>>>(ws, nzero);
    facl_phase1<<<N / 16, 32, 0, stream>>>(R, S, u, v, pos, sum_pos, sum_fair);
    facl_phase2<<<1, 256, 0, stream>>>(u, v, pos, sum_pos, sum_fair,
                                       (float*)d_out, N);
}